// CRFModel_21629455303155
// MI455X (gfx1250) — compile-verified
//
#include <hip/hip_runtime.h>
#include <hip/hip_bf16.h>

#define B_ 64
#define T_ 512
#define E_ 768
#define C_ 20

typedef __attribute__((ext_vector_type(2))) float v2f;
typedef __attribute__((ext_vector_type(8))) float v8f;

// ---------------------------------------------------------------------------
// Kernel 1: emissions[m, c] = hidden[m, :] . fc_w[c, :] + fc_b[c]
// M = B*T = 32768 rows, N = C = 20 (padded to 32), K = E = 768.
// One wave computes a 32x32 tile (2 row-tiles x 2 col-tiles) via
// V_WMMA_F32_16X16X4_F32: 4 loads : 4 WMMAs per K-step of 4.
// hidden is streamed once (non-temporal); fc_w (61 KB) stays cache-resident.
// ---------------------------------------------------------------------------
__global__ __launch_bounds__(256) void gemm_emissions_kernel(
    const float* __restrict__ hidden, const float* __restrict__ fc_w,
    const float* __restrict__ fc_b, float* __restrict__ emis)
{
    const int lane = threadIdx.x & 31;
    const int wave = threadIdx.x >> 5;
    const int tile = blockIdx.x * 8 + wave;     // 32-row tile index
    const int row0 = tile * 32;
    const int half = lane >> 4;                 // 0 or 1
    const int sub  = lane & 15;

    // A (16x4 f32): lane (half,sub) supplies row, K = k+2*half, k+2*half+1
    const float* aptr0 = hidden + (size_t)(row0 + sub)      * E_ + 2 * half;
    const float* aptr1 = hidden + (size_t)(row0 + 16 + sub) * E_ + 2 * half;
    // B (4x16 f32): lane supplies col n, K = k+2*half, k+2*half+1 (B[k][n] = fc_w[n][k])
    const int n0 = sub;
    const int n1 = 16 + sub;
    const float* bptr0 = fc_w + (size_t)n0 * E_ + 2 * half;
    const float* bptr1 = fc_w + (size_t)(n1 < C_ ? n1 : (C_ - 1)) * E_ + 2 * half;

    v8f acc00 = {};   // rows 0-15,  cols 0-15
    v8f acc01 = {};   // rows 0-15,  cols 16-31
    v8f acc10 = {};   // rows 16-31, cols 0-15
    v8f acc11 = {};   // rows 16-31, cols 16-31
    #pragma unroll 4
    for (int k = 0; k < E_; k += 4) {
        v2f a0 = __builtin_nontemporal_load((const v2f*)(aptr0 + k));
        v2f a1 = __builtin_nontemporal_load((const v2f*)(aptr1 + k));
        v2f b0 = *(const v2f*)(bptr0 + k);
        v2f b1 = *(const v2f*)(bptr1 + k);
        acc00 = __builtin_amdgcn_wmma_f32_16x16x4_f32(
            false, a0, false, b0, (short)0, acc00, false, false);
        acc01 = __builtin_amdgcn_wmma_f32_16x16x4_f32(
            false, a0, false, b1, (short)0, acc01, false, false);
        acc10 = __builtin_amdgcn_wmma_f32_16x16x4_f32(
            false, a1, false, b0, (short)0, acc10, false, false);
        acc11 = __builtin_amdgcn_wmma_f32_16x16x4_f32(
            false, a1, false, b1, (short)0, acc11, false, false);
    }

    // C/D layout: VGPR r -> row (r + 8*half), col = sub (+16 for col-tile 1)
    const float bias0 = fc_b[n0];
    const float bias1 = (n1 < C_) ? fc_b[n1] : 0.f;
    #pragma unroll
    for (int r = 0; r < 8; ++r) {
        const int rowA = row0 + r + 8 * half;
        const int rowB = rowA + 16;
        emis[(size_t)rowA * C_ + n0] = acc00[r] + bias0;
        emis[(size_t)rowB * C_ + n0] = acc10[r] + bias0;
        if (n1 < C_) {
            emis[(size_t)rowA * C_ + n1] = acc01[r] + bias1;
            emis[(size_t)rowB * C_ + n1] = acc11[r] + bias1;
        }
    }
}

// ---------------------------------------------------------------------------
// Kernel 2: per-batch CRF. One wave32 per batch row; lane = state.
// Forward logsumexp recurrence + Viterbi max/argmax recurrence + gold-path
// score, all in one pass. Backpointers kept in LDS (u8, 16 KB).
// Single-wave workgroup: LDS ops are in-order, barriers degrade to S_NOP.
// ---------------------------------------------------------------------------
__global__ __launch_bounds__(32) void crf_kernel(
    const float* __restrict__ emis, const int* __restrict__ mask,
    const int* __restrict__ labels, const float* __restrict__ start_t,
    const float* __restrict__ end_t, const float* __restrict__ trans,
    float* __restrict__ llh, float* __restrict__ path_out)
{
    __shared__ float transS[C_ * C_];
    __shared__ float alphaF[32];
    __shared__ float alphaV[32];
    __shared__ float redf[32];
    __shared__ int   redi[32];
    __shared__ unsigned char hist[(T_ - 1) * 32];

    const int b = blockIdx.x;
    const int lane = threadIdx.x;
    const int lc = (lane < C_) ? lane : 0;

    for (int i = lane; i < C_ * C_; i += 32) transS[i] = trans[i];

    float a0 = -1e30f;
    if (lane < C_) a0 = start_t[lane] + emis[(size_t)b * T_ * C_ + lane];
    alphaF[lane] = a0;
    alphaV[lane] = a0;
    __syncthreads();

    // ---- gold-path numerator partial sums (independent terms, strided) ----
    float part = 0.f;
    int msum = 0;
    for (int t = lane; t < T_; t += 32) {
        const int mt = mask[b * T_ + t];
        msum += mt;
        if (t >= 1) {
            const int tc = labels[b * T_ + t];
            const int tp = labels[b * T_ + t - 1];
            part += (transS[tp * C_ + tc] +
                     emis[((size_t)b * T_ + t) * C_ + tc]) * (float)mt;
        }
    }
    redf[lane] = part;
    redi[lane] = msum;
    __syncthreads();

    // ---- sequential recurrence over T ----
    for (int t = 1; t < T_; ++t) {
        const float em = (lane < C_) ? emis[((size_t)b * T_ + t) * C_ + lane] : 0.f;
        const int mt = mask[b * T_ + t];

        float af[C_], av[C_];
        #pragma unroll
        for (int c = 0; c < C_; ++c) {
            const float tr = transS[c * C_ + lc];
            af[c] = alphaF[c] + tr;
            av[c] = alphaV[c] + tr;
        }
        float m = af[0];
        #pragma unroll
        for (int c = 1; c < C_; ++c) m = fmaxf(m, af[c]);
        float s = 0.f;
        #pragma unroll
        for (int c = 0; c < C_; ++c) s += __expf(af[c] - m);
        float bestv = av[0];
        int besti = 0;
        #pragma unroll
        for (int c = 1; c < C_; ++c) {
            if (av[c] > bestv) { bestv = av[c]; besti = c; }
        }
        const float anF = m + __logf(s) + em;
        const float anV = bestv + em;
        __syncthreads();
        if (lane < C_) {
            hist[(t - 1) * 32 + lane] = (unsigned char)besti;
            if (mt) { alphaF[lane] = anF; alphaV[lane] = anV; }
        }
        __syncthreads();
    }

    if (lane == 0) {
        float extra = 0.f;
        int ms = 0;
        for (int i = 0; i < 32; ++i) { extra += redf[i]; ms += redi[i]; }
        const int tag0 = labels[b * T_];
        float score = start_t[tag0] + emis[(size_t)b * T_ * C_ + tag0] + extra;
        const int seq_end = ms - 1;
        score += end_t[labels[b * T_ + seq_end]];

        // log-partition
        float m = -1e30f;
        for (int c = 0; c < C_; ++c) m = fmaxf(m, alphaF[c] + end_t[c]);
        float s = 0.f;
        for (int c = 0; c < C_; ++c) s += __expf(alphaF[c] + end_t[c] - m);
        llh[b] = score - (m + __logf(s));

        // Viterbi terminal + backtrack
        float bv = alphaV[0] + end_t[0];
        int last = 0;
        for (int c = 1; c < C_; ++c) {
            const float v = alphaV[c] + end_t[c];
            if (v > bv) { bv = v; last = c; }
        }
        int tag = last;
        path_out[(size_t)b * T_ + (T_ - 1)] = (float)(last + 1);
        for (int t = T_ - 2; t >= 0; --t) {
            tag = hist[t * 32 + tag];
            path_out[(size_t)b * T_ + t] = (float)(tag + 1);
        }
    }
}

// ---------------------------------------------------------------------------
// Kernel 3: loss = -mean(llh)
// ---------------------------------------------------------------------------
__global__ __launch_bounds__(64) void loss_kernel(const float* __restrict__ llh,
                                                  float* __restrict__ out)
{
    __shared__ float s[64];
    s[threadIdx.x] = llh[threadIdx.x];
    __syncthreads();
    if (threadIdx.x == 0) {
        float a = 0.f;
        for (int i = 0; i < B_; ++i) a += s[i];
        out[0] = -a / (float)B_;
    }
}

extern "C" void kernel_launch(void* const* d_in, const int* in_sizes, int n_in,
                              void* d_out, int out_size, void* d_ws, size_t ws_size,
                              hipStream_t stream) {
    const float* hidden  = (const float*)d_in[0];
    const int*   mask    = (const int*)  d_in[1];
    const int*   labels  = (const int*)  d_in[2];
    const float* fc_w    = (const float*)d_in[3];
    const float* fc_b    = (const float*)d_in[4];
    const float* start_t = (const float*)d_in[5];
    const float* end_t   = (const float*)d_in[6];
    const float* trans   = (const float*)d_in[7];

    float* out  = (float*)d_out;          // [0] = loss, [1..] = path (B*T floats)
    float* emis = (float*)d_ws;           // B*T*C floats = 2.62 MB
    float* llh  = emis + (size_t)B_ * T_ * C_;

    // 32768 rows / 32 per wave = 1024 waves; 8 waves (256 thr) per block.
    gemm_emissions_kernel<<<128, 256, 0, stream>>>(hidden, fc_w, fc_b, emis);
    crf_kernel<<<B_, 32, 0, stream>>>(emis, mask, labels, start_t, end_t, trans,
                                      llh, out + 1);
    loss_kernel<<<1, 64, 0, stream>>>(llh, out);
}